// HaarDWT_61795989455348
// MI455X (gfx1250) — compile-verified
//
#include <hip/hip_runtime.h>
#include <stdint.h>

// ---------------------------------------------------------------------------
// Haar DWT level-1, fp32, bandwidth-bound (~402 MB traffic -> ~17us @ 23.3TB/s).
// CDNA5 streaming path: global_load_async_to_lds_b128 (ASYNCcnt) -> LDS ->
// butterfly in VALU -> non-temporal b128 stores. Wave-private LDS tiles, so no
// workgroup barriers are needed (s_wait_asynccnt is per-wave and in-order).
// ---------------------------------------------------------------------------

typedef float v4f __attribute__((ext_vector_type(4)));

#define IMG_W       1024
#define OUT_W       512
#define NUM_IMG     48                    // 16 batch * 3 channels
#define YH_BASE     (NUM_IMG * OUT_W * OUT_W)   // 12,582,912 floats
#define WAVES_PB    8
#define TILE_COLS   256                   // input columns per wave tile
#define LDS_FLOATS  512                   // 2 rows x 256 cols per wave

#define HAAR_S      0.70710678118654752440f

__global__ __launch_bounds__(256)
void haar_dwt_61795989455348_kernel(const float* __restrict__ in,
                                    float* __restrict__ out)
{
    __shared__ float smem[WAVES_PB * LDS_FLOATS];   // 16 KB

    const uint32_t tid  = threadIdx.x;
    const uint32_t lane = tid & 31u;
    const uint32_t wv   = tid >> 5;                       // wave index in block
    const uint32_t wid  = blockIdx.x * WAVES_PB + wv;     // global wave id

    // tile decode: image q, output row i, column block cb
    const uint32_t q   = wid >> 11;                       // wid / 2048
    const uint32_t rem = wid & 2047u;
    const uint32_t i   = rem >> 2;                        // output row 0..511
    const uint32_t c0  = (rem & 3u) * TILE_COLS;          // input col base

    // ---- async stage: global -> LDS (4 x b128 per lane, tracked on ASYNCcnt)
    const uint32_t row0_b = ((q * IMG_W + 2u * i) * IMG_W + c0) * 4u; // bytes
    const uint32_t row1_b = row0_b + IMG_W * 4u;
    const uint32_t l16    = lane * 16u;

    float* lds = &smem[wv * LDS_FLOATS];
    const uint32_t lds0 = (uint32_t)(uintptr_t)lds;       // LDS byte address

    asm volatile(
        "global_load_async_to_lds_b128 %0, %4, %8\n\t"
        "global_load_async_to_lds_b128 %1, %5, %8\n\t"
        "global_load_async_to_lds_b128 %2, %6, %8\n\t"
        "global_load_async_to_lds_b128 %3, %7, %8"
        :
        : "v"(lds0 + l16),                 // row0, cols [0,128)
          "v"(lds0 + 512u + l16),          // row0, cols [128,256)
          "v"(lds0 + 1024u + l16),         // row1, cols [0,128)
          "v"(lds0 + 1536u + l16),         // row1, cols [128,256)
          "v"(row0_b + l16),
          "v"(row0_b + 512u + l16),
          "v"(row1_b + l16),
          "v"(row1_b + 512u + l16),
          "s"(in)
        : "memory");

    asm volatile("s_wait_asynccnt 0x0" ::: "memory");     // wave-private: no barrier

    // ---- butterfly: lane handles output cols oc0..oc0+3 (input cols 8l..8l+7)
    const v4f* r0 = (const v4f*)lds;              // row 2i
    const v4f* r1 = (const v4f*)(lds + 256);      // row 2i+1

    v4f a0 = r0[2u * lane + 0u];
    v4f a1 = r0[2u * lane + 1u];
    v4f b0 = r1[2u * lane + 0u];
    v4f b1 = r1[2u * lane + 1u];

    v4f p00 = {a0.x, a0.z, a1.x, a1.z};
    v4f p01 = {a0.y, a0.w, a1.y, a1.w};
    v4f p10 = {b0.x, b0.z, b1.x, b1.z};
    v4f p11 = {b0.y, b0.w, b1.y, b1.w};

    // same op order as reference: row pass scaled by S, then column pass by S
    v4f rs0 = HAAR_S * (p00 + p01);   // row_lo, row 2i
    v4f rd0 = HAAR_S * (p00 - p01);   // row_hi, row 2i
    v4f rs1 = HAAR_S * (p10 + p11);   // row_lo, row 2i+1
    v4f rd1 = HAAR_S * (p10 - p11);   // row_hi, row 2i+1

    v4f ll = HAAR_S * (rs0 + rs1);
    v4f lh = HAAR_S * (rs0 - rs1);
    v4f hl = HAAR_S * (rd0 + rd1);
    v4f hh = HAAR_S * (rd0 - rd1);

    // ---- non-temporal, fully coalesced b128 stores (write-once data)
    const uint32_t oc0 = (c0 >> 1) + 4u * lane;
    float* ylp = out + ((q * OUT_W + i) * OUT_W + oc0);
    float* lhp = out + YH_BASE + (((q * 3u + 0u) * OUT_W + i) * OUT_W + oc0);
    float* hlp = out + YH_BASE + (((q * 3u + 1u) * OUT_W + i) * OUT_W + oc0);
    float* hhp = out + YH_BASE + (((q * 3u + 2u) * OUT_W + i) * OUT_W + oc0);

    __builtin_nontemporal_store(ll, (v4f*)ylp);
    __builtin_nontemporal_store(lh, (v4f*)lhp);
    __builtin_nontemporal_store(hl, (v4f*)hlp);
    __builtin_nontemporal_store(hh, (v4f*)hhp);
}

extern "C" void kernel_launch(void* const* d_in, const int* in_sizes, int n_in,
                              void* d_out, int out_size, void* d_ws, size_t ws_size,
                              hipStream_t stream)
{
    const float* x  = (const float*)d_in[0];
    float*       y  = (float*)d_out;

    // 48 images * 512 row-pairs * 4 col-blocks = 98304 wave tiles = 12288 blocks
    dim3 grid(12288), block(256);
    hipLaunchKernelGGL(haar_dwt_61795989455348_kernel, grid, block, 0, stream, x, y);
}